// MLPF_85830626443407
// MI455X (gfx1250) — compile-verified
//
#include <hip/hip_runtime.h>
#include <hip/hip_bf16.h>

#define N_PTS 16384
#define IN_F  34
#define SD    4
#define PD    22
#define KNN   8
#define WID   126
#define WPAD  128
#define NC    6
#define CHUNK 2048

typedef __attribute__((ext_vector_type(2))) float v2f;
typedef __attribute__((ext_vector_type(8))) float v8f;

// ---------------------------------------------------------------------------
// Kernel 1: s = x@Ws + bs + 1000*batch ; h = x@Wh + bh   (conv index 1 only)
// ---------------------------------------------------------------------------
__global__ void __launch_bounds__(256) k_coords(
    const float* __restrict__ x, const int* __restrict__ batch,
    const float* __restrict__ Ws, const float* __restrict__ bs,
    const float* __restrict__ Wh, const float* __restrict__ bh,
    float* __restrict__ s, float* __restrict__ h) {
  __shared__ float wS[IN_F * SD];
  __shared__ float bS[SD];
  __shared__ float wH[IN_F * PD];
  __shared__ float bH[PD];
  for (int t = threadIdx.x; t < IN_F * SD; t += blockDim.x) wS[t] = Ws[t];
  for (int t = threadIdx.x; t < SD; t += blockDim.x) bS[t] = bs[t];
  for (int t = threadIdx.x; t < IN_F * PD; t += blockDim.x) wH[t] = Wh[t];
  for (int t = threadIdx.x; t < PD; t += blockDim.x) bH[t] = bh[t];
  __syncthreads();

  const int i = blockIdx.x * blockDim.x + threadIdx.x;
  float xr[IN_F];
#pragma unroll
  for (int k = 0; k < IN_F; ++k) xr[k] = x[i * IN_F + k];
  const float boff = (float)batch[i] * 1000.0f;

#pragma unroll
  for (int d = 0; d < SD; ++d) {
    float acc = bS[d];
#pragma unroll
    for (int k = 0; k < IN_F; ++k) acc = fmaf(xr[k], wS[k * SD + d], acc);
    s[i * SD + d] = acc + boff;
  }
#pragma unroll
  for (int d = 0; d < PD; ++d) {
    float acc = bH[d];
#pragma unroll
    for (int k = 0; k < IN_F; ++k) acc = fmaf(xr[k], wH[k * PD + d], acc);
    h[i * PD + d] = acc;
  }
}

// ---------------------------------------------------------------------------
// Kernel 2: fused brute-force kNN (full N scan, LDS-tiled) + weighted mean/max
// agg[i] = concat(mean_k(h[idx]*w), max_k(h[idx]*w)), w = exp(-10*d2)
// ---------------------------------------------------------------------------
__global__ void __launch_bounds__(256) k_knn_agg(
    const float* __restrict__ s, const float* __restrict__ h,
    float* __restrict__ agg) {
  __shared__ float4 sc[CHUNK];  // 32 KB candidate tile

  const int i = blockIdx.x * blockDim.x + threadIdx.x;
  const float4* __restrict__ s4 = (const float4*)s;
  const float4 q = s4[i];

  float bd[KNN];
  int   bi[KNN];
#pragma unroll
  for (int t = 0; t < KNN; ++t) { bd[t] = 3.4e38f; bi[t] = 0; }

  for (int c0 = 0; c0 < N_PTS; c0 += CHUNK) {
    __syncthreads();
    for (int t = threadIdx.x; t < CHUNK; t += blockDim.x) sc[t] = s4[c0 + t];
    __syncthreads();

#pragma unroll 4
    for (int j = 0; j < CHUNK; ++j) {
      const float4 p = sc[j];
      const float dx = q.x - p.x, dy = q.y - p.y;
      const float dz = q.z - p.z, dw = q.w - p.w;
      const float d2 = fmaf(dx, dx, fmaf(dy, dy, fmaf(dz, dz, dw * dw)));
      if (d2 < bd[KNN - 1]) {
        const int id = c0 + j;
        // branchless sorted insertion (ascending), static indices -> registers
#pragma unroll
        for (int t = KNN - 1; t >= 1; --t) {
          const bool up  = (bd[t - 1] > d2);
          const bool hit = (bd[t] > d2);
          const float nd = up ? bd[t - 1] : (hit ? d2 : bd[t]);
          const int   ni = up ? bi[t - 1] : (hit ? id : bi[t]);
          bd[t] = nd; bi[t] = ni;
        }
        if (bd[0] > d2) { bd[0] = d2; bi[0] = id; }
      }
    }
  }

  float w[KNN];
#pragma unroll
  for (int t = 0; t < KNN; ++t) w[t] = expf(-10.0f * bd[t]);

  float mn[PD], mx[PD];
#pragma unroll
  for (int c = 0; c < PD; ++c) { mn[c] = 0.0f; mx[c] = -3.4e38f; }
#pragma unroll
  for (int t = 0; t < KNN; ++t) {
    const float* __restrict__ hr = h + (size_t)bi[t] * PD;
#pragma unroll
    for (int c = 0; c < PD; ++c) {
      const float m = hr[c] * w[t];
      mn[c] += m;
      mx[c] = fmaxf(mx[c], m);
    }
  }
  float* __restrict__ ar = agg + (size_t)i * (2 * PD);
#pragma unroll
  for (int c = 0; c < PD; ++c) {
    ar[c]      = mn[c] * (1.0f / (float)KNN);
    ar[PD + c] = mx[c];
  }
}

// ---------------------------------------------------------------------------
// Kernel 3: emb = x@Wo1 + agg@Wo2 + bo2   [N,34]
// ---------------------------------------------------------------------------
__global__ void __launch_bounds__(256) k_embed(
    const float* __restrict__ x, const float* __restrict__ agg,
    const float* __restrict__ Wo1, const float* __restrict__ Wo2,
    const float* __restrict__ bo2, float* __restrict__ emb) {
  __shared__ float w1[IN_F * IN_F];
  __shared__ float w2[2 * PD * IN_F];
  __shared__ float bb[IN_F];
  for (int t = threadIdx.x; t < IN_F * IN_F; t += blockDim.x) w1[t] = Wo1[t];
  for (int t = threadIdx.x; t < 2 * PD * IN_F; t += blockDim.x) w2[t] = Wo2[t];
  for (int t = threadIdx.x; t < IN_F; t += blockDim.x) bb[t] = bo2[t];
  __syncthreads();

  const int i = blockIdx.x * blockDim.x + threadIdx.x;
  float xr[IN_F], ag[2 * PD];
#pragma unroll
  for (int k = 0; k < IN_F; ++k) xr[k] = x[i * IN_F + k];
#pragma unroll
  for (int k = 0; k < 2 * PD; ++k) ag[k] = agg[i * (2 * PD) + k];

#pragma unroll 2
  for (int c = 0; c < IN_F; ++c) {
    float acc = bb[c];
#pragma unroll
    for (int k = 0; k < IN_F; ++k) acc = fmaf(xr[k], w1[k * IN_F + c], acc);
#pragma unroll
    for (int k = 0; k < 2 * PD; ++k) acc = fmaf(ag[k], w2[k * IN_F + c], acc);
    emb[i * IN_F + c] = acc;
  }
}

// ---------------------------------------------------------------------------
// Kernel 4: MLP head via V_WMMA_F32_16X16X4_F32 (fp32 matrix pipe)
// out = elu(elu(emb@W1+b1)@W2+b2)@W3+b3
// One wave per 16-row tile; 4 waves/block; activations staged in LDS.
// ---------------------------------------------------------------------------
__global__ void __launch_bounds__(128) k_head(
    const float* __restrict__ emb,
    const float* __restrict__ W1, const float* __restrict__ b1,
    const float* __restrict__ W2, const float* __restrict__ b2,
    const float* __restrict__ W3, const float* __restrict__ b3,
    float* __restrict__ out) {
  __shared__ float stage[4][16 * WPAD];  // 32 KB: per-wave activation tile

  const int wave = threadIdx.x >> 5;
  const int lane = threadIdx.x & 31;
  const int half = lane >> 4;   // 0: K pair {0,1}, 1: K pair {2,3}
  const int l16  = lane & 15;
  const int r0   = blockIdx.x * 64 + wave * 16;
  float* __restrict__ st = &stage[wave][0];

  // ---------------- Layer 1: 16x34 @ 34x126 (K pad->36, N pad->128) --------
  v8f acc[8];
#pragma unroll
  for (int t = 0; t < 8; ++t) acc[t] = (v8f){};

  const int rowA = r0 + l16;
#pragma unroll
  for (int kk = 0; kk < 9; ++kk) {
    const int ka = kk * 4 + half * 2;
    v2f a;
    a.x = (ka     < IN_F) ? emb[rowA * IN_F + ka]     : 0.0f;
    a.y = (ka + 1 < IN_F) ? emb[rowA * IN_F + ka + 1] : 0.0f;
#pragma unroll
    for (int t = 0; t < 8; ++t) {
      const int col = t * 16 + l16;
      const bool okc = (col < WID);
      v2f b;
      b.x = (ka     < IN_F && okc) ? W1[ka * WID + col]       : 0.0f;
      b.y = (ka + 1 < IN_F && okc) ? W1[(ka + 1) * WID + col] : 0.0f;
      acc[t] = __builtin_amdgcn_wmma_f32_16x16x4_f32(
          false, a, false, b, (short)0, acc[t], false, false);
    }
  }

  // bias + ELU, write h1 tile to LDS (padded cols forced to exact 0)
#pragma unroll
  for (int t = 0; t < 8; ++t) {
    const int col = t * 16 + l16;
    const float bias = (col < WID) ? b1[col] : 0.0f;
#pragma unroll
    for (int r = 0; r < 8; ++r) {
      const int row = half ? (r + 8) : r;
      float v = acc[t][r] + bias;
      v = (v > 0.0f) ? v : expm1f(v);
      if (col >= WID) v = 0.0f;
      st[row * WPAD + col] = v;
    }
  }

  // ---------------- Layer 2: 16x128 @ 128x126 (K=128, N pad->128) ----------
  v8f acc2[8];
#pragma unroll
  for (int t = 0; t < 8; ++t) acc2[t] = (v8f){};

#pragma unroll 1
  for (int kk = 0; kk < 32; ++kk) {
    const int ka = kk * 4 + half * 2;
    v2f a;
    a.x = st[l16 * WPAD + ka];
    a.y = st[l16 * WPAD + ka + 1];
#pragma unroll
    for (int t = 0; t < 8; ++t) {
      const int col = t * 16 + l16;
      const bool okc = (col < WID);
      v2f b;
      b.x = (ka     < WID && okc) ? W2[ka * WID + col]       : 0.0f;
      b.y = (ka + 1 < WID && okc) ? W2[(ka + 1) * WID + col] : 0.0f;
      acc2[t] = __builtin_amdgcn_wmma_f32_16x16x4_f32(
          false, a, false, b, (short)0, acc2[t], false, false);
    }
  }

  // bias + ELU, overwrite LDS tile with h2 (LDS same-wave ops stay in order)
#pragma unroll
  for (int t = 0; t < 8; ++t) {
    const int col = t * 16 + l16;
    const float bias = (col < WID) ? b2[col] : 0.0f;
#pragma unroll
    for (int r = 0; r < 8; ++r) {
      const int row = half ? (r + 8) : r;
      float v = acc2[t][r] + bias;
      v = (v > 0.0f) ? v : expm1f(v);
      if (col >= WID) v = 0.0f;
      st[row * WPAD + col] = v;
    }
  }

  // ---------------- Layer 3: 16x126 @ 126x6 (96 outputs, 3 per lane) -------
#pragma unroll
  for (int j = 0; j < 3; ++j) {
    const int o = lane * 3 + j;       // 0..95
    const int row = o / NC;
    const int col = o % NC;
    float accv = b3[col];
#pragma unroll 2
    for (int k = 0; k < WID; ++k)
      accv = fmaf(st[row * WPAD + k], W3[k * NC + col], accv);
    out[(size_t)(r0 + row) * NC + col] = accv;
  }
}

// ---------------------------------------------------------------------------
extern "C" void kernel_launch(void* const* d_in, const int* in_sizes, int n_in,
                              void* d_out, int out_size, void* d_ws, size_t ws_size,
                              hipStream_t stream) {
  const float* x     = (const float*)d_in[0];
  const int*   batch = (const int*)d_in[1];
  // Only the LAST conv (index 1) reaches the output: each conv reads raw x and
  // the loop overwrites `embedding`, so conv 0 is dead work -> skip it.
  const int ci = 1;
  const float* Ws  = (const float*)d_in[2] + ci * IN_F * SD;
  const float* bs  = (const float*)d_in[3] + ci * SD;
  const float* Wh  = (const float*)d_in[4] + ci * IN_F * PD;
  const float* bh  = (const float*)d_in[5] + ci * PD;
  const float* Wo1 = (const float*)d_in[6] + ci * IN_F * IN_F;
  const float* Wo2 = (const float*)d_in[7] + ci * 2 * PD * IN_F;
  const float* bo2 = (const float*)d_in[8] + ci * IN_F;
  const float* W1  = (const float*)d_in[9];
  const float* b1  = (const float*)d_in[10];
  const float* W2  = (const float*)d_in[11];
  const float* b2  = (const float*)d_in[12];
  const float* W3  = (const float*)d_in[13];
  const float* b3  = (const float*)d_in[14];
  float* out = (float*)d_out;

  float* s   = (float*)d_ws;                    // N*4
  float* h   = s   + (size_t)N_PTS * SD;        // N*22
  float* agg = h   + (size_t)N_PTS * PD;        // N*44
  float* emb = agg + (size_t)N_PTS * 2 * PD;    // N*34   (~6.8 MB total)

  k_coords <<<N_PTS / 256, 256, 0, stream>>>(x, batch, Ws, bs, Wh, bh, s, h);
  k_knn_agg<<<N_PTS / 256, 256, 0, stream>>>(s, h, agg);
  k_embed  <<<N_PTS / 256, 256, 0, stream>>>(x, agg, Wo1, Wo2, bo2, emb);
  k_head   <<<N_PTS / 64, 128, 0, stream>>>(emb, W1, b1, W2, b2, W3, b3, out);
}